// PGNN_layer_90220083020047
// MI455X (gfx1250) — compile-verified
//
#include <hip/hip_runtime.h>
#include <hip/hip_bf16.h>

// ---------------------------------------------------------------------------
// PGNN layer for MI455X (gfx1250, wave32).
//
// Stage 1: u_feat = feature @ Wu + bu ; v_feat = feature @ Wv + bv
//          via V_WMMA_F32_16X16X4_F32 (exact fp32 matrix path).
// Stage 2: per-node edge messages (memory-bound gather stage),
//          out_position[n,k] = relu(u_feat[s]*sp + v_feat[d]) . Wpos + bpos
//          out_structure[n]  = mean_k messages[n,k,:]
// ---------------------------------------------------------------------------

typedef float v2f __attribute__((ext_vector_type(2)));
typedef float v8f __attribute__((ext_vector_type(8)));

#define N_NODES   20000
#define N_ANCHORS 64
#define DIM       128   // D_IN == D_OUT == 128

// ---------------------------------------------------------------------------
// Kernel 1: dual GEMM (feature @ Wu, feature @ Wv) with f32 WMMA.
// grid.x = 20000/16 = 1250 M-tiles; block = 256 threads = 8 waves.
// Wave w computes the 16x16 output tile at columns [16w, 16w+16) for BOTH
// weight matrices, sweeping K in steps of 4 (32 WMMAs per accumulator).
// A-tile (16 rows x 128 cols) staged in LDS, padded to 132 floats/row so the
// column-wise fragment reads hit 16 distinct banks.
// ---------------------------------------------------------------------------
__global__ __launch_bounds__(256) void pgnn_gemm_uv(
    const float* __restrict__ feature,
    const float* __restrict__ Wu, const float* __restrict__ bu,
    const float* __restrict__ Wv, const float* __restrict__ bv,
    float* __restrict__ u_feat, float* __restrict__ v_feat)
{
    __shared__ float Atile[16][132];   // 16x128 tile, padded row stride

    const int m0 = blockIdx.x * 16;
    const int t  = threadIdx.x;

    // Cooperative load of the 16x128 A tile: 2048 floats = 512 float4,
    // 256 threads -> 2 float4 each, fully coalesced.
    #pragma unroll
    for (int i = 0; i < 2; ++i) {
        int idx  = t + i * 256;        // float4 index in tile
        int row  = idx >> 5;           // 32 float4 per row
        int col4 = idx & 31;
        float4 vv = reinterpret_cast<const float4*>(
                        feature + (size_t)(m0 + row) * DIM)[col4];
        Atile[row][col4 * 4 + 0] = vv.x;
        Atile[row][col4 * 4 + 1] = vv.y;
        Atile[row][col4 * 4 + 2] = vv.z;
        Atile[row][col4 * 4 + 3] = vv.w;
    }
    __syncthreads();

    const int lane  = t & 31;
    const int wave  = t >> 5;          // 8 waves -> 8 N-tiles of 16
    const int n0    = wave * 16;
    const int nm    = lane & 15;       // M index for A frag, N index for B frag
    const int khalf = (lane >> 4) * 2; // lanes 0-15 -> K{k,k+1}; 16-31 -> K{k+2,k+3}

    v8f cu = {};
    v8f cv = {};

    for (int k = 0; k < DIM; k += 4) {
        const int kk = k + khalf;
        // A fragment (ISA 7.12.2, 32-bit A 16x4): v0 = A[M][kk], v1 = A[M][kk+1]
        v2f a;
        a.x = Atile[nm][kk + 0];
        a.y = Atile[nm][kk + 1];
        // B fragment (mirrored layout): v0 = W[kk][N], v1 = W[kk+1][N]
        v2f bfu, bfv;
        bfu.x = Wu[(size_t)(kk + 0) * DIM + n0 + nm];
        bfu.y = Wu[(size_t)(kk + 1) * DIM + n0 + nm];
        bfv.x = Wv[(size_t)(kk + 0) * DIM + n0 + nm];
        bfv.y = Wv[(size_t)(kk + 1) * DIM + n0 + nm];

        cu = __builtin_amdgcn_wmma_f32_16x16x4_f32(
                 /*neg_a=*/false, a, /*neg_b=*/false, bfu,
                 /*c_mod=*/(short)0, cu, /*reuse_a=*/false, /*reuse_b=*/false);
        cv = __builtin_amdgcn_wmma_f32_16x16x4_f32(
                 /*neg_a=*/false, a, /*neg_b=*/false, bfv,
                 /*c_mod=*/(short)0, cv, /*reuse_a=*/false, /*reuse_b=*/false);
    }

    // C/D layout: element v of accumulator -> M = v + 8*(lane>=16), N = lane%16
    const float biasu = bu[n0 + nm];
    const float biasv = bv[n0 + nm];
    const int mbase = m0 + ((lane >> 4) << 3);
    #pragma unroll
    for (int v = 0; v < 8; ++v) {
        const size_t off = (size_t)(mbase + v) * DIM + n0 + nm;
        u_feat[off] = cu[v] + biasu;
        v_feat[off] = cv[v] + biasv;
    }
}

// ---------------------------------------------------------------------------
// Kernel 2: edge messages + position head + structure mean.
// One block per node (20000 blocks), 256 threads = 8 waves.
// Wave w handles anchors {w, w+8, ..., w+56}. Each lane owns 4 channels
// (float4), so one wave's gather of a u_feat/v_feat row is exactly one
// coalesced 512 B request -> this kernel runs at the HBM gather roofline
// (~1.31 GB of row gathers ~= 56 us at 23.3 TB/s).
// ---------------------------------------------------------------------------
__global__ __launch_bounds__(256) void pgnn_edges(
    const float* __restrict__ u_feat, const float* __restrict__ v_feat,
    const float* __restrict__ sp_dist, const int* __restrict__ src,
    const int* __restrict__ dst, const int* __restrict__ anchor_eid,
    const float* __restrict__ Wpos, const float* __restrict__ bpos,
    float* __restrict__ out_position, float* __restrict__ out_structure)
{
    const int n    = blockIdx.x;
    const int t    = threadIdx.x;
    const int lane = t & 31;
    const int wave = t >> 5;
    const int c    = lane * 4;          // channel base for this lane

    const float4 wp = *reinterpret_cast<const float4*>(Wpos + c);
    const float  bp = bpos[0];

    float ax = 0.f, ay = 0.f, az = 0.f, aw = 0.f;  // structure accumulators

    __shared__ float4 red[8][32];

    #pragma unroll
    for (int it = 0; it < 8; ++it) {
        const int k  = wave + it * 8;           // anchor index
        const int e  = n * N_ANCHORS + k;       // edge id (uniform per wave)
        const int ae = anchor_eid[e];
        const float sp = sp_dist[ae];
        const int  s  = src[ae];
        const int  d  = dst[ae];

        const float4 u4 = *reinterpret_cast<const float4*>(
                              u_feat + (size_t)s * DIM + c);
        const float4 v4 = *reinterpret_cast<const float4*>(
                              v_feat + (size_t)d * DIM + c);

        const float mx = fmaxf(fmaf(u4.x, sp, v4.x), 0.f);
        const float my = fmaxf(fmaf(u4.y, sp, v4.y), 0.f);
        const float mz = fmaxf(fmaf(u4.z, sp, v4.z), 0.f);
        const float mw = fmaxf(fmaf(u4.w, sp, v4.w), 0.f);

        ax += mx; ay += my; az += mz; aw += mw;

        // position head: dot(message, Wpos) reduced across the wave
        float p = mx * wp.x + my * wp.y + mz * wp.z + mw * wp.w;
        #pragma unroll
        for (int off = 16; off > 0; off >>= 1)
            p += __shfl_xor(p, off, 32);
        if (lane == 0)
            out_position[e] = p + bp;
    }

    red[wave][lane] = make_float4(ax, ay, az, aw);
    __syncthreads();

    // Cross-wave reduce + mean; wave 0 writes the 128-float structure row.
    if (wave == 0) {
        float sx = 0.f, sy = 0.f, sz = 0.f, sw = 0.f;
        #pragma unroll
        for (int w = 0; w < 8; ++w) {
            const float4 r = red[w][lane];
            sx += r.x; sy += r.y; sz += r.z; sw += r.w;
        }
        const float inv = 1.0f / (float)N_ANCHORS;
        float4 o = make_float4(sx * inv, sy * inv, sz * inv, sw * inv);
        *reinterpret_cast<float4*>(out_structure + (size_t)n * DIM + c) = o;
    }
}

// ---------------------------------------------------------------------------
// Launch
// ---------------------------------------------------------------------------
extern "C" void kernel_launch(void* const* d_in, const int* in_sizes, int n_in,
                              void* d_out, int out_size, void* d_ws, size_t ws_size,
                              hipStream_t stream) {
    const float* feature    = (const float*)d_in[0];
    const float* sp_dist    = (const float*)d_in[1];
    const float* Wu         = (const float*)d_in[2];
    const float* bu         = (const float*)d_in[3];
    const float* Wv         = (const float*)d_in[4];
    const float* bv         = (const float*)d_in[5];
    const float* Wpos       = (const float*)d_in[6];
    const float* bpos       = (const float*)d_in[7];
    const int*   src        = (const int*)d_in[8];
    const int*   dst        = (const int*)d_in[9];
    const int*   anchor_eid = (const int*)d_in[10];

    float* u_feat = (float*)d_ws;                               // 20000*128
    float* v_feat = u_feat + (size_t)N_NODES * DIM;             // 20000*128

    float* out_position  = (float*)d_out;                       // 20000*64
    float* out_structure = out_position + (size_t)N_NODES * N_ANCHORS;

    pgnn_gemm_uv<<<N_NODES / 16, 256, 0, stream>>>(
        feature, Wu, bu, Wv, bv, u_feat, v_feat);

    pgnn_edges<<<N_NODES, 256, 0, stream>>>(
        u_feat, v_feat, sp_dist, src, dst, anchor_eid,
        Wpos, bpos, out_position, out_structure);
}